// GCN_Model_29051158790849
// MI455X (gfx1250) — compile-verified
//
#include <hip/hip_runtime.h>

typedef float v2f __attribute__((ext_vector_type(2)));
typedef float v8f __attribute__((ext_vector_type(8)));

#define N_NODES 50000
#define N_EDGES 800000
#define K_DIM   128

// ---------------------------------------------------------------------------
// Dense GEMM  hw[N, M] = h[N, K=128] * W[K=128, M]  using V_WMMA_F32_16X16X4_F32
// Block = 128 threads = 4 waves. Each wave computes one 16-row node tile across
// all M/16 column tiles. W is staged once per block into LDS in a
// pair-interleaved layout so each B fragment is a single ds_load_b64:
//   lds[(k>>1) * 2M + col*2 + (k&1)] = W[k][col]
// ---------------------------------------------------------------------------
template <int M>
__global__ __launch_bounds__(128) void gcn_gemm_wmma(const float* __restrict__ h,
                                                     const float* __restrict__ W,
                                                     float* __restrict__ hw,
                                                     int n_row_tiles) {
  __shared__ float w_lds[K_DIM * M];

  const int tid = threadIdx.x;

  // Cooperative pair-interleaved LDS fill.
  // Each iteration: read float2 from row 2p and row 2p+1 (coalesced),
  // write one interleaved float4 {W[2p][c], W[2p+1][c], W[2p][c+1], W[2p+1][c+1]}.
  {
    constexpr int HALF_M = M / 2;                  // float2 chunks per row
    constexpr int NIT = (K_DIM / 2) * HALF_M;      // total (pair, chunk) items
    for (int i = tid; i < NIT; i += 128) {
      const int p  = i / HALF_M;                   // k-pair index
      const int c2 = (i - p * HALF_M) * 2;         // starting column
      const v2f r0 = *reinterpret_cast<const v2f*>(W + (size_t)(2 * p) * M + c2);
      const v2f r1 = *reinterpret_cast<const v2f*>(W + (size_t)(2 * p + 1) * M + c2);
      float4 packed;
      packed.x = r0.x; packed.y = r1.x;            // col c2:   (k, k+1)
      packed.z = r0.y; packed.w = r1.y;            // col c2+1: (k, k+1)
      *reinterpret_cast<float4*>(&w_lds[p * 2 * M + c2 * 2]) = packed;
    }
  }
  __syncthreads();

  const int wave = tid >> 5;
  const int lane = tid & 31;
  const int row_tile = blockIdx.x * 4 + wave;
  if (row_tile >= n_row_tiles) return;

  const int half = lane >> 4;   // 0 = lanes 0-15, 1 = lanes 16-31
  const int lcol = lane & 15;   // position within half
  const int koff = half * 2;    // K sub-offset per ISA 16x4 f32 A layout

  const int arow = row_tile * 16 + lcol;           // A-matrix row for this lane
  const float* __restrict__ arow_ptr = h + (size_t)arow * K_DIM;

  constexpr int NT = M / 16;    // column tiles (8 for M=128, 4 for M=64)
  v8f acc[NT];
#pragma unroll
  for (int t = 0; t < NT; ++t) acc[t] = (v8f)0.0f;

#pragma unroll 4
  for (int k = 0; k < K_DIM; k += 4) {
    // A fragment: lanes 0-15: A[row][k..k+1], lanes 16-31: A[row][k+2..k+3]
    v2f a = *reinterpret_cast<const v2f*>(arow_ptr + k + koff);
    // Pair row in LDS for this half: (k + koff)/2 = k/2 + half
    const int pbase = ((k >> 1) + half) * 2 * M;
#pragma unroll
    for (int t = 0; t < NT; ++t) {
      // Single ds_load_b64: {W[k+koff][col], W[k+koff+1][col]}
      v2f b = *reinterpret_cast<const v2f*>(&w_lds[pbase + (t * 16 + lcol) * 2]);
      acc[t] = __builtin_amdgcn_wmma_f32_16x16x4_f32(
          /*neg_a=*/false, a, /*neg_b=*/false, b,
          /*c_mod=*/(short)0, acc[t], /*reuse_a=*/false, /*reuse_b=*/false);
    }
  }

  // C/D layout: VGPR v -> row = tile*16 + v + 8*half, col = tile_n*16 + lcol
  const int out_row0 = row_tile * 16 + half * 8;
#pragma unroll
  for (int t = 0; t < NT; ++t) {
#pragma unroll
    for (int v = 0; v < 8; ++v) {
      hw[(size_t)(out_row0 + v) * M + t * 16 + lcol] = acc[t][v];
    }
  }
}

// ---------------------------------------------------------------------------
// out[n, c] = bias[c]  (so the scatter can accumulate straight on top)
// ---------------------------------------------------------------------------
template <int M>
__global__ void gcn_bias_init(float* __restrict__ out, const float* __restrict__ bias) {
  constexpr int CPR = M / 4;                       // float4 chunks per row
  const int i = blockIdx.x * blockDim.x + threadIdx.x;
  if (i >= N_NODES * CPR) return;
  const int c = (i & (CPR - 1));                   // chunk within row
  const float4 b = reinterpret_cast<const float4*>(bias)[c];
  reinterpret_cast<float4*>(out)[i] = b;
}

// ---------------------------------------------------------------------------
// Scatter-add: out[dst[e], :] += hw[src[e], :]
// One edge per M/4 consecutive lanes -> coalesced 512B row reads, hardware
// fp32 atomics (global_atomic_add_f32) for accumulation.
// ---------------------------------------------------------------------------
template <int M>
__global__ void gcn_scatter(const float* __restrict__ hw,
                            const int* __restrict__ src,
                            const int* __restrict__ dst,
                            float* __restrict__ out) {
  constexpr int CPE = M / 4;                       // float4 chunks per edge
  constexpr int SH = (M == 128) ? 5 : 4;           // log2(CPE)
  const long long t = (long long)blockIdx.x * blockDim.x + threadIdx.x;
  const long long e = t >> SH;
  if (e >= N_EDGES) return;
  const int c = (int)(t & (CPE - 1)) * 4;

  const int s = src[e];
  const int d = dst[e];
  const float4 v = *reinterpret_cast<const float4*>(hw + (size_t)s * M + c);
  float* o = out + (size_t)d * M + c;
  unsafeAtomicAdd(o + 0, v.x);
  unsafeAtomicAdd(o + 1, v.y);
  unsafeAtomicAdd(o + 2, v.z);
  unsafeAtomicAdd(o + 3, v.w);
}

// ---------------------------------------------------------------------------
extern "C" void kernel_launch(void* const* d_in, const int* in_sizes, int n_in,
                              void* d_out, int out_size, void* d_ws, size_t ws_size,
                              hipStream_t stream) {
  (void)in_sizes; (void)n_in; (void)out_size; (void)ws_size;

  const float* x  = (const float*)d_in[0];
  const int*   ei = (const int*)d_in[1];
  const float* W1 = (const float*)d_in[2];
  const float* b1 = (const float*)d_in[3];
  const float* W2 = (const float*)d_in[4];
  const float* b2 = (const float*)d_in[5];
  const float* W3 = (const float*)d_in[6];
  const float* b3 = (const float*)d_in[7];

  const int* src = ei;             // edge_index[0]
  const int* dst = ei + N_EDGES;   // edge_index[1]

  float* o1 = (float*)d_out;
  float* o2 = o1 + (size_t)N_NODES * 128;
  float* o3 = o2 + (size_t)N_NODES * 128;
  float* hw = (float*)d_ws;        // [N_NODES, 128] fp32 scratch (25.6 MB)

  const int n_row_tiles = N_NODES / 16;            // 3125 (exact)
  const dim3 gemm_grid((n_row_tiles + 3) / 4);     // 4 waves/block, 1 tile/wave

  const int bias_blocks_128 = (N_NODES * (128 / 4) + 255) / 256;
  const int bias_blocks_64  = (N_NODES * (64 / 4) + 255) / 256;
  const int scat_blocks_128 = (int)(((long long)N_EDGES * 32 + 255) / 256);
  const int scat_blocks_64  = (int)(((long long)N_EDGES * 16 + 255) / 256);

  // Layer 1: o1 = scatter(x @ W1) + b1
  gcn_gemm_wmma<128><<<gemm_grid, 128, 0, stream>>>(x, W1, hw, n_row_tiles);
  gcn_bias_init<128><<<bias_blocks_128, 256, 0, stream>>>(o1, b1);
  gcn_scatter<128><<<scat_blocks_128, 256, 0, stream>>>(hw, src, dst, o1);

  // Layer 2: o2 = scatter(o1 @ W2) + b2
  gcn_gemm_wmma<128><<<gemm_grid, 128, 0, stream>>>(o1, W2, hw, n_row_tiles);
  gcn_bias_init<128><<<bias_blocks_128, 256, 0, stream>>>(o2, b2);
  gcn_scatter<128><<<scat_blocks_128, 256, 0, stream>>>(hw, src, dst, o2);

  // Layer 3: o3 = scatter(o2 @ W3) + b3   (D_OUT = 64)
  gcn_gemm_wmma<64><<<gemm_grid, 128, 0, stream>>>(o2, W3, hw, n_row_tiles);
  gcn_bias_init<64><<<bias_blocks_64, 256, 0, stream>>>(o3, b3);
  gcn_scatter<64><<<scat_blocks_64, 256, 0, stream>>>(hw, src, dst, o3);
}